// Target_67465346285899
// MI455X (gfx1250) — compile-verified
//
#include <hip/hip_runtime.h>
#include <cstring>
#include <math.h>

typedef __attribute__((ext_vector_type(2))) float v2f;
typedef __attribute__((ext_vector_type(8))) float v8f;

#define NHEADS 4
#define NEG_SLOPE 0.2f

// ---------- helpers: order-preserving float<->uint for atomicMax ----------
__device__ __forceinline__ unsigned f2ord(float f) {
    unsigned u = __float_as_uint(f);
    return (u & 0x80000000u) ? ~u : (u | 0x80000000u);
}
__device__ __forceinline__ float ord2f(unsigned u) {
    return (u & 0x80000000u) ? __uint_as_float(u & 0x7FFFFFFFu) : __uint_as_float(~u);
}

// =======================================================================
// WMMA f32 GEMM: Out[N,M] = X[N,K] @ W[K,M], optional (acc+bias)*scale, relu
// Each wave computes a 16 x (16*CTILES) tile using V_WMMA_F32_16X16X4_F32.
// A-frag 16x4 layout (ISA 7.12.2): lanes 0-15 row=lane K={0,1}, lanes 16-31
// row=lane-16 K={2,3}. B-frag 4x16 mirrors (rows split low/high lane half).
// C/D 16x16: col = lane&15, row = 8*(lane>>4) + vgpr_index.
// =======================================================================
template <int CTILES>
__global__ void gemm_wmma_kernel(const float* __restrict__ X,
                                 const float* __restrict__ W,
                                 float* __restrict__ Out,
                                 int N, int K, int M,
                                 const float* __restrict__ bias,
                                 float scale, int do_relu) {
    const int lane = threadIdx.x & 31;
    const long wave = ((long)blockIdx.x * blockDim.x + threadIdx.x) >> 5;
    const int colGroups = M / (16 * CTILES);
    const int rowTiles = (N + 15) >> 4;
    if (wave >= (long)rowTiles * colGroups) return;
    const int rowTile = (int)(wave / colGroups);
    const int cg = (int)(wave % colGroups);
    const int row0 = rowTile << 4;
    const int col0 = cg * (16 * CTILES);
    const int half = lane >> 4;   // 0: K=0,1 / rows 0-7 ; 1: K=2,3 / rows 8-15
    const int l = lane & 15;

    v8f acc[CTILES];
#pragma unroll
    for (int t = 0; t < CTILES; ++t) {
        v8f z = {0.f, 0.f, 0.f, 0.f, 0.f, 0.f, 0.f, 0.f};
        acc[t] = z;
    }

    int arow = row0 + l;
    if (arow >= N) arow = N - 1;  // N is a multiple of 16 here; safety clamp
    const float* __restrict__ xr = X + (long)arow * K;

    for (int k0 = 0; k0 < K; k0 += 4) {
        const int ka = k0 + (half << 1);
        v2f a;
        a.x = xr[ka];
        a.y = xr[ka + 1];
#pragma unroll
        for (int t = 0; t < CTILES; ++t) {
            const int col = col0 + t * 16 + l;
            v2f b;
            b.x = W[(long)ka * M + col];
            b.y = W[(long)(ka + 1) * M + col];
            acc[t] = __builtin_amdgcn_wmma_f32_16x16x4_f32(
                false, a, false, b, (short)0, acc[t], false, false);
        }
    }

#pragma unroll
    for (int t = 0; t < CTILES; ++t) {
        const int col = col0 + t * 16 + l;
        const float bv = bias ? bias[col] : 0.f;
#pragma unroll
        for (int r = 0; r < 8; ++r) {
            const int row = row0 + half * 8 + r;
            if (row < N) {
                float v = (acc[t][r] + bv) * scale;
                if (do_relu) v = fmaxf(v, 0.f);
                Out[(long)row * M + col] = v;
            }
        }
    }
}

// ---------- per-(node,head) attention score dots ----------
__global__ void att_scores_kernel(const float* __restrict__ H,
                                  const float* __restrict__ As,
                                  const float* __restrict__ Ad,
                                  float* __restrict__ a_s,
                                  float* __restrict__ a_d,
                                  int N, int C, int HC) {
    const long wave = ((long)blockIdx.x * blockDim.x + threadIdx.x) >> 5;
    const int lane = threadIdx.x & 31;
    if (wave >= (long)N * NHEADS) return;
    const int n = (int)(wave >> 2);
    const int h = (int)(wave & 3);
    const float* hp = H + (long)n * HC + h * C;
    const float* asp = As + h * C;
    const float* adp = Ad + h * C;
    float s = 0.f, d = 0.f;
    for (int c = lane; c < C; c += 32) {
        float v = hp[c];
        s += v * asp[c];
        d += v * adp[c];
    }
#pragma unroll
    for (int o = 16; o > 0; o >>= 1) {
        s += __shfl_xor(s, o, 32);
        d += __shfl_xor(d, o, 32);
    }
    if (lane == 0) {
        a_s[(long)n * NHEADS + h] = s;
        a_d[(long)n * NHEADS + h] = d;
    }
}

// ---------- fill uint pattern ----------
__global__ void fill_u32_kernel(unsigned* __restrict__ p, unsigned v, int n) {
    int i = blockIdx.x * blockDim.x + threadIdx.x;
    if (i < n) p[i] = v;
}

// ---------- edge pass 1: leaky-relu logit + segment max ----------
__global__ void edge_logits_max_kernel(const int* __restrict__ ei, int E, int Etot,
                                       const float* __restrict__ a_s,
                                       const float* __restrict__ a_d,
                                       float* __restrict__ wbuf,
                                       unsigned* __restrict__ nmax) {
    int e = blockIdx.x * blockDim.x + threadIdx.x;
    if (e >= Etot) return;
    int s, d;
    if (e < E) { s = ei[e]; d = ei[E + e]; } else { s = d = e - E; }
#pragma unroll
    for (int h = 0; h < NHEADS; ++h) {
        float x = a_s[(long)s * NHEADS + h] + a_d[(long)d * NHEADS + h];
        x = (x > 0.f) ? x : NEG_SLOPE * x;
        wbuf[(long)e * NHEADS + h] = x;
        atomicMax(&nmax[(long)d * NHEADS + h], f2ord(x));
    }
}

// ---------- edge pass 2: exp(logit - max) + segment sum ----------
__global__ void edge_exp_denom_kernel(const int* __restrict__ ei, int E, int Etot,
                                      float* __restrict__ wbuf,
                                      const unsigned* __restrict__ nmax,
                                      float* __restrict__ den) {
    int e = blockIdx.x * blockDim.x + threadIdx.x;
    if (e >= Etot) return;
    int d = (e < E) ? ei[E + e] : (e - E);
#pragma unroll
    for (int h = 0; h < NHEADS; ++h) {
        float m = ord2f(nmax[(long)d * NHEADS + h]);
        float w = expf(wbuf[(long)e * NHEADS + h] - m);
        wbuf[(long)e * NHEADS + h] = w;
        atomicAdd(&den[(long)d * NHEADS + h], w);
    }
}

// ---------- edge pass 3: wave-per-edge gather * alpha -> scatter add ----------
__global__ void edge_aggregate_kernel(const int* __restrict__ ei, int E, int Etot,
                                      const float* __restrict__ wbuf,
                                      const float* __restrict__ den,
                                      const float* __restrict__ H,
                                      float* __restrict__ acc,
                                      int cshift, int HC) {
    const long wave = ((long)blockIdx.x * blockDim.x + threadIdx.x) >> 5;
    const int lane = threadIdx.x & 31;
    if (wave >= Etot) return;
    const int e = (int)wave;
    int s, d;
    if (e < E) { s = ei[e]; d = ei[E + e]; } else { s = d = e - E; }
    const float* hp = H + (long)s * HC;
    float* ap = acc + (long)d * HC;
    for (int i = lane; i < HC; i += 32) {
        const int h = i >> cshift;
        const float alpha = wbuf[(long)e * NHEADS + h] /
                            (den[(long)d * NHEADS + h] + 1e-16f);
        atomicAdd(&ap[i], alpha * hp[i]);
    }
}

// ---------- head mean + bias (+relu) ----------
__global__ void finalize_kernel(const float* __restrict__ acc,
                                const float* __restrict__ bias,
                                float* __restrict__ out,
                                int N, int C, int HC, int do_relu) {
    long idx = (long)blockIdx.x * blockDim.x + threadIdx.x;
    if (idx >= (long)N * C) return;
    const int n = (int)(idx / C);
    const int c = (int)(idx % C);
    const float* ap = acc + (long)n * HC;
    float v = 0.f;
#pragma unroll
    for (int h = 0; h < NHEADS; ++h) v += ap[h * C + c];
    v = v * (1.f / NHEADS) + bias[c];
    if (do_relu) v = fmaxf(v, 0.f);
    out[idx] = v;
}

// ---------- tiny classifier: [N,16] @ [16,40] + bias ----------
__global__ void mlp2_kernel(const float* __restrict__ T,
                            const float* __restrict__ Wm2,
                            const float* __restrict__ bm2,
                            float* __restrict__ out, int N, int NC) {
    long idx = (long)blockIdx.x * blockDim.x + threadIdx.x;
    if (idx >= (long)N * NC) return;
    const int n = (int)(idx / NC);
    const int j = (int)(idx % NC);
    float v = bm2[j];
#pragma unroll
    for (int k = 0; k < 16; ++k) v += T[(long)n * 16 + k] * Wm2[k * NC + j];
    out[idx] = v;
}

// =======================================================================
// Host-side orchestration
// =======================================================================
static inline long cdiv(long a, long b) { return (a + b - 1) / b; }

static void launch_gemm(const float* X, const float* W, float* Out,
                        int N, int K, int M,
                        const float* bias, float scale, int relu,
                        hipStream_t s) {
    const long rowTiles = cdiv(N, 16);
    if (M % 64 == 0) {
        const long waves = rowTiles * (M / 64);
        gemm_wmma_kernel<4><<<(unsigned)cdiv(waves, 8), 256, 0, s>>>(
            X, W, Out, N, K, M, bias, scale, relu);
    } else {
        const long waves = rowTiles * (M / 16);
        gemm_wmma_kernel<1><<<(unsigned)cdiv(waves, 8), 256, 0, s>>>(
            X, W, Out, N, K, M, bias, scale, relu);
    }
}

static void run_attention(const int* ei, int E, int Etot, int N,
                          const float* H, const float* As, const float* Ad,
                          const float* bias,
                          float* a_s, float* a_d, float* den, unsigned* nmax,
                          float* wbuf, float* acc, float* out,
                          int C, int HC, int cshift, int do_relu,
                          unsigned ordneg, hipStream_t stream) {
    const long scoreWaves = (long)N * NHEADS;
    att_scores_kernel<<<(unsigned)cdiv(scoreWaves, 8), 256, 0, stream>>>(
        H, As, Ad, a_s, a_d, N, C, HC);
    hipMemsetAsync(den, 0, (size_t)N * NHEADS * sizeof(float), stream);
    hipMemsetAsync(acc, 0, (size_t)N * HC * sizeof(float), stream);
    fill_u32_kernel<<<(unsigned)cdiv((long)N * NHEADS, 256), 256, 0, stream>>>(
        nmax, ordneg, N * NHEADS);
    edge_logits_max_kernel<<<(unsigned)cdiv(Etot, 256), 256, 0, stream>>>(
        ei, E, Etot, a_s, a_d, wbuf, nmax);
    edge_exp_denom_kernel<<<(unsigned)cdiv(Etot, 256), 256, 0, stream>>>(
        ei, E, Etot, wbuf, nmax, den);
    edge_aggregate_kernel<<<(unsigned)cdiv(Etot, 8), 256, 0, stream>>>(
        ei, E, Etot, wbuf, den, H, acc, cshift, HC);
    finalize_kernel<<<(unsigned)cdiv((long)N * C, 256), 256, 0, stream>>>(
        acc, bias, out, N, C, HC, do_relu);
}

extern "C" void kernel_launch(void* const* d_in, const int* in_sizes, int n_in,
                              void* d_out, int out_size, void* d_ws, size_t ws_size,
                              hipStream_t stream) {
    const float* x   = (const float*)d_in[0];
    const int*   ei  = (const int*)d_in[1];
    const float* W1  = (const float*)d_in[2];
    const float* As1 = (const float*)d_in[3];
    const float* Ad1 = (const float*)d_in[4];
    const float* b1  = (const float*)d_in[5];
    const float* W2  = (const float*)d_in[6];
    const float* As2 = (const float*)d_in[7];
    const float* Ad2 = (const float*)d_in[8];
    const float* b2  = (const float*)d_in[9];
    const float* W3  = (const float*)d_in[10];
    const float* As3 = (const float*)d_in[11];
    const float* Ad3 = (const float*)d_in[12];
    const float* b3  = (const float*)d_in[13];
    const float* Wm1 = (const float*)d_in[14];
    const float* bm1 = (const float*)d_in[15];
    const float* Wm2 = (const float*)d_in[16];
    const float* bm2 = (const float*)d_in[17];

    const int IN = 256, HID = 64, OUT = 128, NC = 40;
    const int N = in_sizes[0] / IN;
    const int E = in_sizes[1] / 2;
    const int Etot = E + N;

    // workspace layout (fp32 elements)
    float* BUF_H   = (float*)d_ws;                       // N*512 (max proj width)
    float* BUF_ACC = BUF_H + (size_t)N * 512;            // N*512
    float* BUF_Z   = BUF_ACC + (size_t)N * 512;          // N*128 (layer out / mlp hidden)
    float* a_s     = BUF_Z + (size_t)N * 128;            // N*4
    float* a_d     = a_s + (size_t)N * NHEADS;           // N*4
    float* den     = a_d + (size_t)N * NHEADS;           // N*4
    unsigned* nmax = (unsigned*)(den + (size_t)N * NHEADS); // N*4
    float* wbuf    = (float*)(nmax + (size_t)N * NHEADS);   // Etot*4

    float* out_cls = (float*)d_out;                 // [N,40]
    float* x3      = out_cls + (size_t)N * NC;      // [N,128]

    // encoded -inf sentinel for the float atomicMax trick
    float neg = -3.0e38f;
    unsigned nu;
    memcpy(&nu, &neg, 4);
    const unsigned ordneg = (nu & 0x80000000u) ? ~nu : (nu | 0x80000000u);

    // ---- layer 1: 256 -> 4x64 ----
    launch_gemm(x, W1, BUF_H, N, IN, NHEADS * HID, nullptr, 1.f, 0, stream);
    run_attention(ei, E, Etot, N, BUF_H, As1, Ad1, b1, a_s, a_d, den, nmax,
                  wbuf, BUF_ACC, BUF_Z, HID, NHEADS * HID, 6, /*relu=*/1,
                  ordneg, stream);

    // ---- layer 2: 64 -> 4x64 ----
    launch_gemm(BUF_Z, W2, BUF_H, N, HID, NHEADS * HID, nullptr, 1.f, 0, stream);
    run_attention(ei, E, Etot, N, BUF_H, As2, Ad2, b2, a_s, a_d, den, nmax,
                  wbuf, BUF_ACC, BUF_Z, HID, NHEADS * HID, 6, /*relu=*/1,
                  ordneg, stream);

    // ---- layer 3: 64 -> 4x128, output x3 (no relu) ----
    launch_gemm(BUF_Z, W3, BUF_H, N, HID, NHEADS * OUT, nullptr, 1.f, 0, stream);
    run_attention(ei, E, Etot, N, BUF_H, As3, Ad3, b3, a_s, a_d, den, nmax,
                  wbuf, BUF_ACC, x3, OUT, NHEADS * OUT, 7, /*relu=*/0,
                  ordneg, stream);

    // ---- MLP head: (x3 @ Wm1 + bm1) * BN_SCALE, relu  ->  @ Wm2 + bm2 ----
    const float bnscale = 1.0f / sqrtf(1.0f + 1e-5f);
    launch_gemm(x3, Wm1, BUF_Z, N, OUT, 16, bm1, bnscale, /*relu=*/1, stream);
    mlp2_kernel<<<(unsigned)cdiv((long)N * NC, 256), 256, 0, stream>>>(
        BUF_Z, Wm2, bm2, out_cls, N, NC);
}